// LightGCN_21157008900739
// MI455X (gfx1250) — compile-verified
//
#include <hip/hip_runtime.h>

typedef __attribute__((ext_vector_type(2))) float v2f;
typedef __attribute__((ext_vector_type(4))) float v4f;
typedef __attribute__((ext_vector_type(8))) float v8f;

#define EMB 64

// ---- init: x0 = acc = concat(user_emb, item_emb) ----
// ue/ie read once -> NT; x0 will be gathered by layer-0 SpMM -> RT store;
// acc not needed until the accumulate pass -> NT store (don't pollute L2).
__global__ void lg_init4(const v4f* __restrict__ ue, const v4f* __restrict__ ie,
                         v4f* __restrict__ x0, v4f* __restrict__ acc,
                         int userVec, int totalVec) {
    int i = blockIdx.x * blockDim.x + threadIdx.x;
    if (i >= totalVec) return;
    v4f v = (i < userVec) ? __builtin_nontemporal_load(ue + i)
                          : __builtin_nontemporal_load(ie + (i - userVec));
    x0[i] = v;                                   // RT: layer-0 gather source
    __builtin_nontemporal_store(v, acc + i);     // NT: accumulator
}

// RT stores: ydst must be L2-resident for the atomic scatter that follows.
__global__ void lg_zero4(v4f* __restrict__ p, int nVec) {
    int i = blockIdx.x * blockDim.x + threadIdx.x;
    if (i < nVec) p[i] = (v4f){0.f, 0.f, 0.f, 0.f};
}

// ---- SpMM: one edge per 16 lanes, float4 gather, f32 atomic scatter ----
// Edge lists are pure streams (345 MB over 3 layers): NT loads + prefetch so
// they never evict the 154 MB embedding working set from the 192 MB L2.
__global__ void lg_spmm(const int* __restrict__ rows, const int* __restrict__ cols,
                        const float* __restrict__ vals, const float* __restrict__ x,
                        float* __restrict__ y, int nEdges) {
    int tid = blockIdx.x * blockDim.x + threadIdx.x;
    int e = tid >> 4;          // 16 lanes per edge (64 dims / 4 per lane)
    if (e >= nEdges) return;
    int q = tid & 15;
    if (q == 0) {              // stream edge lists ahead (global_prefetch_b8)
        __builtin_prefetch(rows + e + 4096, 0, 0);
        __builtin_prefetch(cols + e + 4096, 0, 0);
        __builtin_prefetch(vals + e + 4096, 0, 0);
    }
    int   r = __builtin_nontemporal_load(rows + e);
    int   c = __builtin_nontemporal_load(cols + e);
    float v = __builtin_nontemporal_load(vals + e);
    const v4f g = *reinterpret_cast<const v4f*>(x + (size_t)c * EMB + 4 * q); // RT: L2-hot
    float* dst = y + (size_t)r * EMB + 4 * q;
    atomicAdd(dst + 0, v * g.x);
    atomicAdd(dst + 1, v * g.y);
    atomicAdd(dst + 2, v * g.z);
    atomicAdd(dst + 3, v * g.w);
}

// ---- acc += y via V_WMMA_F32_16X16X4_F32 (D = I*B + C), one wave per 16x16 tile ----
// acc traffic is touch-once-per-pass -> NT both directions; y stays RT (it is
// the next layer's gather source).
__global__ void lg_wmma_acc(const float* __restrict__ y, float* __restrict__ acc,
                            int nTileRows) {
    int wavesPerBlock = blockDim.x >> 5;
    int waveId = blockIdx.x * wavesPerBlock + (threadIdx.x >> 5);
    int totalTiles = nTileRows * (EMB / 16);
    if (waveId >= totalTiles) return;          // wave-uniform: EXEC stays all-1s for WMMA
    int lane = threadIdx.x & 31;
    int n    = lane & 15;
    int half = lane >> 4;

    int tr = waveId >> 2;                      // tile row (16 matrix rows)
    int tc = waveId & 3;                       // tile col (16 of 64 dims)
    int rowBase = tr * 16;
    int colBase = tc * 16 + n;

    // C: 16x16 f32 accumulator tile, VGPR r -> rows r (lanes 0-15) / r+8 (lanes 16-31)
    v8f c;
#pragma unroll
    for (int r = 0; r < 8; ++r)
        c[r] = __builtin_nontemporal_load(acc + (size_t)(rowBase + r + 8 * half) * EMB + colBase);

    v8f d = c;
#pragma unroll
    for (int k = 0; k < 4; ++k) {
        // A_k = columns 4k..4k+3 of the 16x16 identity (16x4, 2 VGPRs)
        int j0 = 4 * k + 2 * half;             // this lane holds K = j0 (x) and j0+1 (y)
        v2f a, b;
        a.x = (n == j0)     ? 1.0f : 0.0f;
        a.y = (n == j0 + 1) ? 1.0f : 0.0f;
        // B_k = rows 4k..4k+3 of the Y tile (4x16, 2 VGPRs, K positions mirror A)
        b.x = y[(size_t)(rowBase + j0)     * EMB + colBase];
        b.y = y[(size_t)(rowBase + j0 + 1) * EMB + colBase];
        d = __builtin_amdgcn_wmma_f32_16x16x4_f32(
                /*neg_a=*/false, a, /*neg_b=*/false, b,
                /*c_mod=*/(short)0, d, /*reuse_a=*/false, /*reuse_b=*/false);
    }

#pragma unroll
    for (int r = 0; r < 8; ++r)
        __builtin_nontemporal_store(d[r], acc + (size_t)(rowBase + r + 8 * half) * EMB + colBase);
}

// ---- scalar tail accumulate (unused when N % 16 == 0; N=300000 is divisible) ----
__global__ void lg_add4(const v4f* __restrict__ y, v4f* __restrict__ acc, int nVec) {
    int i = blockIdx.x * blockDim.x + threadIdx.x;
    if (i >= nVec) return;
    v4f a = __builtin_nontemporal_load(acc + i);
    v4f b = y[i];
    a += b;
    __builtin_nontemporal_store(a, acc + i);
}

__global__ void lg_scale4(v4f* __restrict__ p, float s, int nVec) {
    int i = blockIdx.x * blockDim.x + threadIdx.x;
    if (i >= nVec) return;
    v4f a = __builtin_nontemporal_load(p + i);
    a *= s;
    __builtin_nontemporal_store(a, p + i);
}

extern "C" void kernel_launch(void* const* d_in, const int* in_sizes, int n_in,
                              void* d_out, int out_size, void* d_ws, size_t ws_size,
                              hipStream_t stream) {
    const float* ue  = (const float*)d_in[0];
    const float* ie  = (const float*)d_in[1];
    const int*   rows = (const int*)d_in[2];
    const int*   cols = (const int*)d_in[3];
    const float* vals = (const float*)d_in[4];

    const int userElems  = in_sizes[0];           // NUM_USERS * 64
    const int itemElems  = in_sizes[1];           // NUM_ITEMS * 64
    const int E          = in_sizes[2];           // edges
    const int totalElems = userElems + itemElems; // N * 64
    const int N          = totalElems / EMB;

    float* acc = (float*)d_out;                   // accumulator lives in d_out
    float* ws0 = (float*)d_ws;                    // ping
    float* ws1 = ws0 + (size_t)totalElems;        // pong

    const int B = 256;                            // 8 wave32 per block
    const int totalVec = totalElems / 4;
    const int userVec  = userElems / 4;
    const int vecBlocks = (totalVec + B - 1) / B;

    lg_init4<<<vecBlocks, B, 0, stream>>>((const v4f*)ue, (const v4f*)ie,
                                          (v4f*)ws0, (v4f*)acc, userVec, totalVec);

    const int nTileRows = N / 16;
    const int tailRows  = N - nTileRows * 16;
    const int tiles     = nTileRows * (EMB / 16);
    const int wavesPerBlock = B / 32;
    const int accBlocks = (tiles + wavesPerBlock - 1) / wavesPerBlock;

    const long spThreads = (long)E * 16;
    const int  spBlocks  = (int)((spThreads + B - 1) / B);

    float* xsrc = ws0;
    float* ydst = ws1;
    for (int layer = 0; layer < 3; ++layer) {
        lg_zero4<<<vecBlocks, B, 0, stream>>>((v4f*)ydst, totalVec);
        lg_spmm<<<spBlocks, B, 0, stream>>>(rows, cols, vals, xsrc, ydst, E);
        if (tiles > 0)
            lg_wmma_acc<<<accBlocks, B, 0, stream>>>(ydst, acc, nTileRows);
        if (tailRows > 0) {
            const int tailVec = tailRows * EMB / 4;
            const size_t off  = (size_t)nTileRows * 16 * EMB;
            lg_add4<<<(tailVec + B - 1) / B, B, 0, stream>>>(
                (const v4f*)(ydst + off), (v4f*)(acc + off), tailVec);
        }
        float* t = xsrc; xsrc = ydst; ydst = t;
    }

    lg_scale4<<<vecBlocks, B, 0, stream>>>((v4f*)acc, 0.25f, totalVec);
}